// GNN_Critic_23476291240204
// MI455X (gfx1250) — compile-verified
//
#include <hip/hip_runtime.h>

// CDNA5 / gfx1250. wave32. WMMA f32 16x16x4 for the per-node 1->16 linear.
typedef __attribute__((ext_vector_type(2))) float v2f;
typedef __attribute__((ext_vector_type(8))) float v8f;

// ds_swizzle_b32 butterfly helper (group-of-32 mode, lane' = lane ^ XOR):
// offset = {1'b0, xor[4:0], or[4:0]=0, and[4:0]=0x1f}
template <int IMM>
__device__ __forceinline__ float swz(float v) {
  return __int_as_float(__builtin_amdgcn_ds_swizzle(__float_as_int(v), IMM));
}
#define SWAPX1 0x041f
#define SWAPX2 0x081f
#define SWAPX4 0x101f
#define SWAPX8 0x201f

// ---------------------------------------------------------------------------
// K0: zero accumulators (sum1, deg, sum2 contiguous at ws[0..3N)) and out.
// ---------------------------------------------------------------------------
__global__ void k_zero(float* __restrict__ ws, float* __restrict__ out,
                       int nws, int nout) {
  int i = blockIdx.x * blockDim.x + threadIdx.x;
  if (i < nws)             ws[i] = 0.0f;
  else if (i - nws < nout) out[i - nws] = 0.0f;
}

// ---------------------------------------------------------------------------
// K1: layer-1 edge pass, 4 edges/thread with b128 index loads:
//     sum1[dst] += x[src]; deg[dst] += 1.
// ---------------------------------------------------------------------------
__global__ void k_edge1_v4(const int4* __restrict__ srcv,
                           const int4* __restrict__ dstv,
                           const float* __restrict__ x,
                           float* __restrict__ sum1, float* __restrict__ deg,
                           int nvec) {
  int i = blockIdx.x * blockDim.x + threadIdx.x;
  if (i >= nvec) return;
  __builtin_prefetch(srcv + i + 2048, 0, 1);   // global_prefetch_b8, speculative
  __builtin_prefetch(dstv + i + 2048, 0, 1);
  int4 s4 = srcv[i];                           // global_load_b128
  int4 d4 = dstv[i];
  float x0 = x[s4.x], x1 = x[s4.y], x2 = x[s4.z], x3 = x[s4.w];  // L2-resident
  unsafeAtomicAdd(&sum1[d4.x], x0);  unsafeAtomicAdd(&deg[d4.x], 1.0f);
  unsafeAtomicAdd(&sum1[d4.y], x1);  unsafeAtomicAdd(&deg[d4.y], 1.0f);
  unsafeAtomicAdd(&sum1[d4.z], x2);  unsafeAtomicAdd(&deg[d4.z], 1.0f);
  unsafeAtomicAdd(&sum1[d4.w], x3);  unsafeAtomicAdd(&deg[d4.w], 1.0f);
}

__global__ void k_edge1_tail(const int* __restrict__ ei, const float* __restrict__ x,
                             float* __restrict__ sum1, float* __restrict__ deg,
                             int start, int E) {
  int e = start + blockIdx.x * blockDim.x + threadIdx.x;
  if (e >= E) return;
  int src = ei[e];
  int dst = ei[E + e];
  unsafeAtomicAdd(&sum1[dst], x[src]);
  unsafeAtomicAdd(&deg[dst], 1.0f);
}

// ---------------------------------------------------------------------------
// K2: per-node pass via WMMA. One wave handles 16 nodes with a single
// v_wmma_f32_16x16x4_f32:
//   A[m,0..3] = [mean1[node_m], x[node_m], 0, 0]      (M=16, K=4)
//   B[0..3,n] = [W1l[n]; W1r[n]; 0; 0]                (K=4,  N=16)
//   C[m,n]    = b1[n]
//   D[m,n]    = pre-ReLU hidden h[node_m][n]
// ReLU in the C/D layout, then the two 16-dim dots (s=h.W2l, t=h.W2r) via a
// 16-lane ds_swizzle XOR butterfly (masks 1,2,4,8 keep lane halves separate).
// ---------------------------------------------------------------------------
__global__ void k_node_wmma(const float* __restrict__ x,
                            const float* __restrict__ sum1,
                            const float* __restrict__ deg,
                            float* __restrict__ invd,
                            float* __restrict__ s_out,
                            float* __restrict__ t_out,
                            const float* __restrict__ W1l,
                            const float* __restrict__ b1,
                            const float* __restrict__ W1r,
                            const float* __restrict__ W2l,
                            const float* __restrict__ W2r,
                            int ntiles) {
  int wid  = threadIdx.x >> 5;
  int tile = blockIdx.x * (blockDim.x >> 5) + wid;
  if (tile >= ntiles) return;              // wave-uniform exit (EXEC stays full)
  int lane = threadIdx.x & 31;
  int half = lane >> 4;                    // 0: K=0,1 / M=0..7 ; 1: K=2,3 / M=8..15
  int f    = lane & 15;                    // feature (N) index; node-in-tile for A
  int base = tile * 16;

  float m1 = 0.0f, xv = 0.0f;
  if (half == 0) {
    int node = base + f;
    float iv = 1.0f / fmaxf(deg[node], 1.0f);
    invd[node] = iv;                       // reused by the final pass
    m1 = sum1[node] * iv;                  // mean_j x_j
    xv = x[node];
  }
  v2f a; a[0] = m1; a[1] = xv;             // K=2,3 rows (half==1) are zero

  float w1l = W1l[f], w1r = W1r[f];
  v2f b;
  b[0] = (half == 0) ? w1l : 0.0f;
  b[1] = (half == 0) ? w1r : 0.0f;

  float bias = b1[f];
  v8f c;
#pragma unroll
  for (int r = 0; r < 8; ++r) c[r] = bias; // C[m][n] = b1[n]

  v8f d = __builtin_amdgcn_wmma_f32_16x16x4_f32(
      /*neg_a=*/false, a, /*neg_b=*/false, b,
      /*c_mod=*/(short)0, c, /*reuse_a=*/false, /*reuse_b=*/false);

  float w2l = W2l[f], w2r = W2r[f];
  float p[8], q[8];
#pragma unroll
  for (int r = 0; r < 8; ++r) {
    float h = fmaxf(d[r], 0.0f);           // ReLU
    p[r] = h * w2l;
    q[r] = h * w2r;
  }
  // butterfly sum over N within each 16-lane half (immediate ds_swizzle)
#pragma unroll
  for (int r = 0; r < 8; ++r) {
    p[r] += swz<SWAPX1>(p[r]);  q[r] += swz<SWAPX1>(q[r]);
    p[r] += swz<SWAPX2>(p[r]);  q[r] += swz<SWAPX2>(q[r]);
    p[r] += swz<SWAPX4>(p[r]);  q[r] += swz<SWAPX4>(q[r]);
    p[r] += swz<SWAPX8>(p[r]);  q[r] += swz<SWAPX8>(q[r]);
  }
  if (f == 0) {                            // lane 0 -> M=0..7, lane 16 -> M=8..15
    int nb = base + (half ? 8 : 0);
#pragma unroll
    for (int r = 0; r < 8; ++r) {
      s_out[nb + r] = p[r];
      t_out[nb + r] = q[r];
    }
  }
}

// scalar fallback for N % 16 nodes (unused for N=200000, kept for generality)
__global__ void k_node_tail(const float* __restrict__ x,
                            const float* __restrict__ sum1,
                            const float* __restrict__ deg,
                            float* __restrict__ invd,
                            float* __restrict__ s_out,
                            float* __restrict__ t_out,
                            const float* __restrict__ W1l,
                            const float* __restrict__ b1,
                            const float* __restrict__ W1r,
                            const float* __restrict__ W2l,
                            const float* __restrict__ W2r,
                            int start, int N) {
  int n = start + blockIdx.x * blockDim.x + threadIdx.x;
  if (n >= N) return;
  float iv = 1.0f / fmaxf(deg[n], 1.0f);
  invd[n] = iv;
  float m1 = sum1[n] * iv, xv = x[n];
  float s = 0.0f, t = 0.0f;
#pragma unroll
  for (int k = 0; k < 16; ++k) {
    float h = fmaxf(fmaf(m1, W1l[k], fmaf(xv, W1r[k], b1[k])), 0.0f);
    s = fmaf(h, W2l[k], s);
    t = fmaf(h, W2r[k], t);
  }
  s_out[n] = s; t_out[n] = t;
}

// ---------------------------------------------------------------------------
// K3: layer-2 edge pass, collapsed to SCALAR by linearity of lin_l:
//     W2l @ mean_j(h_j) = mean_j(h_j . W2l) = mean_j s_j
// ---------------------------------------------------------------------------
__global__ void k_edge2_v4(const int4* __restrict__ srcv,
                           const int4* __restrict__ dstv,
                           const float* __restrict__ s,
                           float* __restrict__ sum2, int nvec) {
  int i = blockIdx.x * blockDim.x + threadIdx.x;
  if (i >= nvec) return;
  __builtin_prefetch(srcv + i + 2048, 0, 1);
  __builtin_prefetch(dstv + i + 2048, 0, 1);
  int4 s4 = srcv[i];
  int4 d4 = dstv[i];
  float v0 = s[s4.x], v1 = s[s4.y], v2 = s[s4.z], v3 = s[s4.w];
  unsafeAtomicAdd(&sum2[d4.x], v0);
  unsafeAtomicAdd(&sum2[d4.y], v1);
  unsafeAtomicAdd(&sum2[d4.z], v2);
  unsafeAtomicAdd(&sum2[d4.w], v3);
}

__global__ void k_edge2_tail(const int* __restrict__ ei, const float* __restrict__ s,
                             float* __restrict__ sum2, int start, int E) {
  int e = start + blockIdx.x * blockDim.x + threadIdx.x;
  if (e >= E) return;
  unsafeAtomicAdd(&sum2[ei[E + e]], s[ei[e]]);
}

// ---------------------------------------------------------------------------
// K4: finalize node value and global_add_pool by graph id.
//     h2[n] = sum2[n]*invd[n] + b2 + t[n];  out[batch[n]] += h2[n]
// ---------------------------------------------------------------------------
__global__ void k_final(const float* __restrict__ sum2,
                        const float* __restrict__ invd,
                        const float* __restrict__ t,
                        const int* __restrict__ batch,
                        const float* __restrict__ b2,
                        float* __restrict__ out, int N) {
  int n = blockIdx.x * blockDim.x + threadIdx.x;
  if (n >= N) return;
  float h2 = fmaf(sum2[n], invd[n], b2[0] + t[n]);
  unsafeAtomicAdd(&out[batch[n]], h2);
}

// ---------------------------------------------------------------------------
extern "C" void kernel_launch(void* const* d_in, const int* in_sizes, int n_in,
                              void* d_out, int out_size, void* d_ws, size_t ws_size,
                              hipStream_t stream) {
  const float* x     = (const float*)d_in[0];
  const int*   ei    = (const int*)d_in[1];   // [2, E]
  const int*   batch = (const int*)d_in[2];   // [N], sorted
  // d_in[3] = batch_size scalar (== out_size)
  const float* W1l = (const float*)d_in[4];   // [16]
  const float* b1  = (const float*)d_in[5];   // [16]
  const float* W1r = (const float*)d_in[6];   // [16]
  const float* W2l = (const float*)d_in[7];   // [16]
  const float* b2  = (const float*)d_in[8];   // [1]
  const float* W2r = (const float*)d_in[9];   // [16]

  int N = in_sizes[0];
  int E = in_sizes[1] / 2;

  float* ws   = (float*)d_ws;
  float* sum1 = ws;            // [N]  zeroed
  float* deg  = ws + N;        // [N]  zeroed
  float* sum2 = ws + 2 * N;    // [N]  zeroed
  float* invd = ws + 3 * N;    // [N]  fully overwritten
  float* sv   = ws + 4 * N;    // [N]  fully overwritten
  float* tv   = ws + 5 * N;    // [N]  fully overwritten
  float* out  = (float*)d_out; // [B]

  int zlen = 3 * N + out_size;
  k_zero<<<(zlen + 255) / 256, 256, 0, stream>>>(ws, out, 3 * N, out_size);

  // layer-1 edge pass (vectorized when E%4==0 keeps ei+E 16B-aligned)
  int nvec = E / 4;
  const int4* srcv = (const int4*)ei;
  const int4* dstv = (const int4*)(ei + E);
  bool vec_ok = (E % 4) == 0;
  if (vec_ok) {
    k_edge1_v4<<<(nvec + 255) / 256, 256, 0, stream>>>(srcv, dstv, x, sum1, deg, nvec);
  } else {
    k_edge1_tail<<<(E + 255) / 256, 256, 0, stream>>>(ei, x, sum1, deg, 0, E);
  }

  int ntiles = N / 16;
  if (ntiles > 0)
    k_node_wmma<<<(ntiles + 7) / 8, 256, 0, stream>>>(
        x, sum1, deg, invd, sv, tv, W1l, b1, W1r, W2l, W2r, ntiles);
  int tail = N - ntiles * 16;
  if (tail > 0)
    k_node_tail<<<1, 32, 0, stream>>>(
        x, sum1, deg, invd, sv, tv, W1l, b1, W1r, W2l, W2r, ntiles * 16, N);

  // layer-2 edge pass (scalar messages)
  if (vec_ok) {
    k_edge2_v4<<<(nvec + 255) / 256, 256, 0, stream>>>(srcv, dstv, sv, sum2, nvec);
  } else {
    k_edge2_tail<<<(E + 255) / 256, 256, 0, stream>>>(ei, sv, sum2, 0, E);
  }

  k_final<<<(N + 255) / 256, 256, 0, stream>>>(sum2, invd, tv, batch, b2, out, N);
}